// L_CA1_88648124991342
// MI455X (gfx1250) — compile-verified
//
#include <hip/hip_runtime.h>

typedef __attribute__((ext_vector_type(2))) float v2f;
typedef __attribute__((ext_vector_type(8))) float v8f;

#define BATCH    8192
#define N_ITEMS  4096
#define N_CA3    2048
#define N_CA1    64
#define KWTA     6
#define TAU      0.1f

#define THREADS        128   // 4 wave32s
#define WAVES          4
#define ROWS_PER_BLOCK 64    // 16 rows per wave
#define SROW           66    // padded LDS row stride (floats): conflict-free scatter

// One wave computes a 16-row x 64-col C strip as 4 accumulators (n-tiles 0..3),
// reusing a single A fragment per k-step across 4 V_WMMA_F32_16X16X4_F32.
// A  is [M][K] row-major; Bt is the pre-transposed weights [64][K] row-major,
// so both A and B fragments are contiguous per-lane b64 streams along K:
//   A frag: lane (l&15) = row M,  VGPR pair holds K = k + 2*(l>>4), +1
//   B frag: lane (l&15) = col N,  VGPR pair holds K = k + 2*(l>>4), +1
__device__ __forceinline__ void wmma_gemm4(const float* __restrict__ A,
                                           const float* __restrict__ Bt,
                                           int K, int m0, int lane,
                                           v8f (&c)[4]) {
  const int row = lane & 15;
  const int kb  = (lane >> 4) << 1;  // 0 or 2
  const float* __restrict__ arow = A  + (size_t)(m0 + row) * (size_t)K + kb;
  const float* __restrict__ b0   = Bt + (size_t)(row)      * (size_t)K + kb;
  const float* __restrict__ b1   = b0 + (size_t)16 * K;
  const float* __restrict__ b2   = b0 + (size_t)32 * K;
  const float* __restrict__ b3   = b0 + (size_t)48 * K;

  for (int k0 = 0; k0 < K; k0 += 64) {
    __builtin_prefetch(arow + k0 + 256, 0, 1);  // stream A ~1KB ahead (Bt is L2-hot)
#pragma unroll
    for (int k = 0; k < 64; k += 4) {
      v2f a  = *(const v2f*)(arow + k0 + k);
      v2f v0 = *(const v2f*)(b0   + k0 + k);
      v2f v1 = *(const v2f*)(b1   + k0 + k);
      v2f v2 = *(const v2f*)(b2   + k0 + k);
      v2f v3 = *(const v2f*)(b3   + k0 + k);
      // (neg_a, A, neg_b, B, c_mod, C, reuse_a, reuse_b)
      c[0] = __builtin_amdgcn_wmma_f32_16x16x4_f32(false, a, false, v0, (short)0, c[0], false, false);
      c[1] = __builtin_amdgcn_wmma_f32_16x16x4_f32(false, a, false, v1, (short)0, c[1], false, false);
      c[2] = __builtin_amdgcn_wmma_f32_16x16x4_f32(false, a, false, v2, (short)0, c[2], false, false);
      c[3] = __builtin_amdgcn_wmma_f32_16x16x4_f32(false, a, false, v3, (short)0, c[3], false, false);
    }
  }
}

// Transpose W[K][64] (row-major) into Wt[64][K]; coalesced reads, tiny volume.
__global__ __launch_bounds__(256) void transpose_w_kernel(
    const float* __restrict__ w, float* __restrict__ wt, int K) {
  int idx = blockIdx.x * 256 + threadIdx.x;  // over K*64 elements
  int k = idx >> 6;
  int n = idx & 63;
  wt[(size_t)n * (size_t)K + k] = w[idx];
}

__global__ __launch_bounds__(THREADS) void ca1_fused_kernel(
    const float* __restrict__ aECin, const float* __restrict__ aCA3,
    const float* __restrict__ aECout,
    const float* __restrict__ wtECin, const float* __restrict__ wtCA3,
    const float* __restrict__ wtECout,
    const float* __restrict__ activity,
    float* __restrict__ out) {
  __shared__ float sact[ROWS_PER_BLOCK * SROW];  // padded act tile (~16.5 KB)
  __shared__ float skth[ROWS_PER_BLOCK];         // per-row kWTA threshold

  const int tid  = threadIdx.x;
  const int lane = tid & 31;
  const int wave = tid >> 5;
  const int m0   = blockIdx.x * ROWS_PER_BLOCK + wave * 16;

  // --- fused 3-way GEMM, fp32 WMMA, fp32 accumulate ---
  v8f c[4] = {v8f{}, v8f{}, v8f{}, v8f{}};
  wmma_gemm4(aECin,  wtECin,  N_ITEMS, m0, lane, c);
  wmma_gemm4(aCA3,   wtCA3,   N_CA3,   m0, lane, c);
  wmma_gemm4(aECout, wtECout, N_ITEMS, m0, lane, c);

  // --- nxx1 activation, scatter C fragments to LDS ---
  // C/D layout: lane (l&15) = col N within tile; VGPR i -> row M = i + 8*(l>>4)
  const int nlo = lane & 15;
  const int mb  = wave * 16 + ((lane >> 4) << 3);
#pragma unroll
  for (int t = 0; t < 4; ++t) {
#pragma unroll
    for (int i = 0; i < 8; ++i) {
      float pos = fmaxf(c[t][i], 0.0f);
      sact[(mb + i) * SROW + t * 16 + nlo] = pos / (pos + 1.0f);
    }
  }
  __syncthreads();

  // --- exact top-6-of-64 per row (one thread per row) ---
  if (tid < ROWS_PER_BLOCK) {
    float top[KWTA];
#pragma unroll
    for (int s = 0; s < KWTA; ++s) top[s] = -1.0f;  // act >= 0 always
    const float* rowv = &sact[tid * SROW];
    for (int j = 0; j < N_CA1; ++j) {
      float v = rowv[j];
      if (v > top[KWTA - 1]) {
        top[KWTA - 1] = v;
#pragma unroll
        for (int s = KWTA - 1; s > 0; --s) {
          if (top[s] > top[s - 1]) {
            float t2 = top[s]; top[s] = top[s - 1]; top[s - 1] = t2;
          }
        }
      }
    }
    skth[tid] = top[KWTA - 1];  // k-th largest (dup-aware, matches top_k)
  }
  __syncthreads();

  // --- kWTA mask + Euler step, fully coalesced global I/O ---
#pragma unroll 4
  for (int j = 0; j < (ROWS_PER_BLOCK * N_CA1) / THREADS; ++j) {
    int e = tid + j * THREADS;          // 0..4095 within block tile
    int r = e >> 6;
    int col = e & 63;
    float av   = sact[r * SROW + col];
    float kth  = skth[r];
    float sp   = (av >= kth) ? av : 0.0f;
    size_t g   = (size_t)blockIdx.x * (ROWS_PER_BLOCK * N_CA1) + e;
    float act0 = activity[g];
    out[g] = act0 + TAU * (sp - act0);
  }
}

extern "C" void kernel_launch(void* const* d_in, const int* in_sizes, int n_in,
                              void* d_out, int out_size, void* d_ws, size_t ws_size,
                              hipStream_t stream) {
  (void)in_sizes; (void)n_in; (void)out_size; (void)ws_size;
  const float* aECin    = (const float*)d_in[0];
  const float* aCA3     = (const float*)d_in[1];
  const float* aECout   = (const float*)d_in[2];
  const float* wECin    = (const float*)d_in[3];
  const float* wCA3     = (const float*)d_in[4];
  const float* wECout   = (const float*)d_in[5];
  const float* activity = (const float*)d_in[6];
  float* out = (float*)d_out;

  // Workspace layout: transposed weights Wt[64][K], back-to-back.
  float* wtECin  = (float*)d_ws;
  float* wtCA3   = wtECin + (size_t)N_ITEMS * N_CA1;
  float* wtECout = wtCA3  + (size_t)N_CA3   * N_CA1;

  // Prep: transpose the three weight matrices (2.6 MB total, lands in L2).
  hipLaunchKernelGGL(transpose_w_kernel, dim3((N_ITEMS * N_CA1) / 256), dim3(256),
                     0, stream, wECin, wtECin, N_ITEMS);
  hipLaunchKernelGGL(transpose_w_kernel, dim3((N_CA3 * N_CA1) / 256), dim3(256),
                     0, stream, wCA3, wtCA3, N_CA3);
  hipLaunchKernelGGL(transpose_w_kernel, dim3((N_ITEMS * N_CA1) / 256), dim3(256),
                     0, stream, wECout, wtECout, N_ITEMS);

  // Main fused kernel: 128 blocks x 128 threads (4 wave32s, 64 rows/block).
  hipLaunchKernelGGL(ca1_fused_kernel, dim3(BATCH / ROWS_PER_BLOCK), dim3(THREADS),
                     0, stream,
                     aECin, aCA3, aECout, wtECin, wtCA3, wtECout, activity, out);
}